// CannyWrapper_32384053412195
// MI455X (gfx1250) — compile-verified
//
#include <hip/hip_runtime.h>
#include <cstdint>

#define H_IMG 4096
#define W_IMG 4096
#define WPR   128   // 32-bit mask words per image row

// ---------------- CDNA5 async global->LDS path (gfx1250) ----------------
#if __has_builtin(__builtin_amdgcn_global_load_async_to_lds_b32)
#define USE_ASYNC_LDS 1
#else
#define USE_ASYNC_LDS 0
#endif

__device__ __forceinline__ void async_g2l_b32(const void* g, void* l) {
#if USE_ASYNC_LDS
  __builtin_amdgcn_global_load_async_to_lds_b32(
      (__attribute__((address_space(1))) int*)(uintptr_t)g,
      (__attribute__((address_space(3))) int*)(unsigned)(uintptr_t)l,
      0, 0);
#else
  *(unsigned*)l = *(const unsigned*)g;
#endif
}

__device__ __forceinline__ void wait_async_lds() {
#if USE_ASYNC_LDS
#if __has_builtin(__builtin_amdgcn_s_wait_asynccnt)
  __builtin_amdgcn_s_wait_asynccnt(0);
#else
  asm volatile("s_wait_asynccnt 0" ::: "memory");
#endif
#endif
}

__device__ __forceinline__ unsigned ballot32(bool p) {
#if __has_builtin(__builtin_amdgcn_ballot_w32)
  return __builtin_amdgcn_ballot_w32(p);
#else
  return (unsigned)__ballot((int)p);
#endif
}

// ---------------- Kernel 1: Sobel + NMS + threshold, bit-packed ----------------
// block (64,4) = 256 threads = 8 wave32; each wave = 32 consecutive x of one row.
// tile: 64 wide x 32 tall.  grid (64,128).
__global__ void __launch_bounds__(256)
k_sobel_nms(const float* __restrict__ img, unsigned* __restrict__ weak,
            unsigned* __restrict__ strong)
{
  __shared__ float in_s[36][72];   // input tile, halo 2 (replicate border)
  __shared__ float mag_s[34][68];  // mag tile, halo 1 (zero outside image)

  const int x0 = blockIdx.x * 64;
  const int y0 = blockIdx.y * 32;
  const int tid  = threadIdx.y * 64 + threadIdx.x;
  const int lane = tid & 31;
  const int wave = tid >> 5;

  // stage 68x36 fp32 tile into LDS via async DMA; no div/mod index math
  for (int r = wave; r < 36; r += 8) {
    int gy = y0 + r - 2; gy = gy < 0 ? 0 : (gy > H_IMG - 1 ? H_IMG - 1 : gy);
    const float* rowp = img + (size_t)gy * W_IMG;
    for (int c = lane; c < 68; c += 32) {
      int gx = x0 + c - 2; gx = gx < 0 ? 0 : (gx > W_IMG - 1 ? W_IMG - 1 : gx);
      async_g2l_b32(rowp + gx, &in_s[r][c]);
    }
  }
  wait_async_lds();
  __syncthreads();

  // Sobel L1 magnitude on (66 x 34) region; zero for out-of-image coords
  for (int r = wave; r < 34; r += 8) {
    const int yy = y0 + r - 1;
    const bool yin = (yy >= 0) && (yy < H_IMG);
    for (int c = lane; c < 66; c += 32) {
      int xx = x0 + c - 1;
      float m = 0.0f;
      if (yin && xx >= 0 && xx < W_IMG) {
        float a00 = in_s[r][c],     a01 = in_s[r][c + 1],     a02 = in_s[r][c + 2];
        float a10 = in_s[r + 1][c],                           a12 = in_s[r + 1][c + 2];
        float a20 = in_s[r + 2][c], a21 = in_s[r + 2][c + 1], a22 = in_s[r + 2][c + 2];
        float gxv = (a02 + 2.0f * a12 + a22) - (a00 + 2.0f * a10 + a20);
        float gyv = (a20 + 2.0f * a21 + a22) - (a00 + 2.0f * a01 + a02);
        m = fabsf(gxv) + fabsf(gyv);
      }
      mag_s[r][c] = m;
    }
  }
  __syncthreads();

  const int lx = threadIdx.x;  // 0..63
  for (int i = 0; i < 8; ++i) {
    const int ly = threadIdx.y + 4 * i;  // 0..31
    // gradient direction at center (input centered at in_s[ly+2][lx+2])
    float a00 = in_s[ly + 1][lx + 1], a01 = in_s[ly + 1][lx + 2], a02 = in_s[ly + 1][lx + 3];
    float a10 = in_s[ly + 2][lx + 1],                              a12 = in_s[ly + 2][lx + 3];
    float a20 = in_s[ly + 3][lx + 1], a21 = in_s[ly + 3][lx + 2], a22 = in_s[ly + 3][lx + 3];
    float gxv = (a02 + 2.0f * a12 + a22) - (a00 + 2.0f * a10 + a20);
    float gyv = (a20 + 2.0f * a21 + a22) - (a00 + 2.0f * a01 + a02);
    float ax = fabsf(gxv), ay = fabsf(gyv);
    float m = mag_s[ly + 1][lx + 1];
    bool horiz = ay < ax * 0.41421356237309503f;  // tan(22.5)
    bool vert  = ay > ax * 2.4142135623730951f;   // tan(67.5)
    bool ss    = (gxv * gyv) >= 0.0f;
    float n1 = horiz ? mag_s[ly + 1][lx]
             : (vert ? mag_s[ly][lx + 1]
             : (ss   ? mag_s[ly][lx] : mag_s[ly][lx + 2]));
    float n2 = horiz ? mag_s[ly + 1][lx + 2]
             : (vert ? mag_s[ly + 2][lx + 1]
             : (ss   ? mag_s[ly + 2][lx + 2] : mag_s[ly + 2][lx]));
    bool keep = (m > 100.0f) && (m > n1) && (m >= n2);
    bool strg = keep && (m > 200.0f);
    unsigned kw = ballot32(keep);   // wave32: 32 consecutive x -> one mask word
    unsigned sw = ballot32(strg);
    if ((lx & 31) == 0) {
      int widx = (y0 + ly) * WPR + (x0 >> 5) + (lx >> 5);
      weak[widx]   = kw;
      strong[widx] = sw;
    }
  }
}

// ---------------- Kernel 2: bit-packed hysteresis propagation ----------------
// block (8,32): thread owns one mask word (32 px). tile 256 px x 32 rows.
// Ping-pong buffers -> each launch is a pure deterministic function of ein.
__global__ void __launch_bounds__(256)
k_hyst(const unsigned* __restrict__ weak, const unsigned* __restrict__ ein,
       unsigned* __restrict__ eout, int inner)
{
  __shared__ unsigned e_s[34][10];  // edges tile + 1-row/1-word halo
  __shared__ unsigned w_s[32][8];   // weak, interior only

  const int wc0 = blockIdx.x * 8;
  const int r0  = blockIdx.y * 32;
  const int tid = threadIdx.y * 8 + threadIdx.x;

  // haloed edge tile: in-range words via async DMA, out-of-range -> 0
  for (int i = tid; i < 34 * 10; i += 256) {
    int r = i / 10, c = i % 10;
    int gr = r0 + r - 1, gc = wc0 + c - 1;
    if (gr >= 0 && gr < H_IMG && gc >= 0 && gc < WPR)
      async_g2l_b32(&ein[gr * WPR + gc], &e_s[r][c]);
    else
      e_s[r][c] = 0u;
  }
  async_g2l_b32(&weak[(r0 + threadIdx.y) * WPR + wc0 + threadIdx.x],
                &w_s[threadIdx.y][threadIdx.x]);
  wait_async_lds();
  __syncthreads();

  const int r = threadIdx.y + 1, c = threadIdx.x + 1;
  const unsigned wk = w_s[threadIdx.y][threadIdx.x];
  for (int k = 0; k < inner; ++k) {
    unsigned up  = e_s[r - 1][c], ul = e_s[r - 1][c - 1], ur = e_s[r - 1][c + 1];
    unsigned mid = e_s[r][c],     ml = e_s[r][c - 1],     mr = e_s[r][c + 1];
    unsigned dn  = e_s[r + 1][c], dl = e_s[r + 1][c - 1], dr = e_s[r + 1][c + 1];
    // bit i = pixel x; <<1 spreads to x+1, >>1 to x-1; cross-word via bit 31/0
    unsigned hu = up | (up << 1) | (up >> 1) | (ul >> 31) | (ur << 31);
    unsigned hm = (mid << 1) | (mid >> 1) | (ml >> 31) | (mr << 31);
    unsigned hd = dn | (dn << 1) | (dn >> 1) | (dl >> 31) | (dr << 31);
    unsigned nw = mid | (wk & (hu | hm | hd));
    int changed = (nw != mid);
    if (!__syncthreads_or(changed)) break;  // tile-local fixed point
    e_s[r][c] = nw;
    __syncthreads();
  }
  eout[(r0 + threadIdx.y) * WPR + wc0 + threadIdx.x] = e_s[r][c];
}

// ---------------- Kernel 3: expand bits -> fp32 0/255, 16B stores ----------------
__global__ void __launch_bounds__(256)
k_final(const unsigned* __restrict__ edges, float4* __restrict__ out)
{
  int i = blockIdx.x * 256 + threadIdx.x;  // quad-pixel index
  unsigned w = edges[i >> 3];              // 8 quads per mask word, L2-resident
  int sh = (i & 7) * 4;
  float4 v;
  v.x = ((w >> (sh + 0)) & 1u) ? 255.0f : 0.0f;
  v.y = ((w >> (sh + 1)) & 1u) ? 255.0f : 0.0f;
  v.z = ((w >> (sh + 2)) & 1u) ? 255.0f : 0.0f;
  v.w = ((w >> (sh + 3)) & 1u) ? 255.0f : 0.0f;
  out[i] = v;
}

// ---------------- Host launch ----------------
extern "C" void kernel_launch(void* const* d_in, const int* in_sizes, int n_in,
                              void* d_out, int out_size, void* d_ws, size_t ws_size,
                              hipStream_t stream) {
  const float* img = (const float*)d_in[0];
  float4* out = (float4*)d_out;

  const size_t NW = (size_t)H_IMG * WPR;  // 524288 words = 2 MB
  unsigned* weak = (unsigned*)d_ws;
  unsigned* e0 = weak + NW;
  unsigned* e1 = e0 + NW;

  k_sobel_nms<<<dim3(W_IMG / 64, H_IMG / 32), dim3(64, 4), 0, stream>>>(img, weak, e0);

  unsigned* bufs[2] = {e0, e1};
  const int PASSES = 16;  // 16 x (<=24 in-LDS steps + cross-tile halo exchange)
  for (int p = 0; p < PASSES; ++p) {
    k_hyst<<<dim3(WPR / 8, H_IMG / 32), dim3(8, 32), 0, stream>>>(
        weak, bufs[p & 1], bufs[(p + 1) & 1], 24);
  }
  // PASSES even -> final result back in e0
  k_final<<<dim3((H_IMG * W_IMG / 4) / 256), 256, 0, stream>>>(bufs[PASSES & 1], out);
}